// KnowledgeBasedLoss_19610820673649
// MI455X (gfx1250) — compile-verified
//
#include <hip/hip_runtime.h>
#include <math.h>

// ---------------------------------------------------------------------------
// KnowledgeBasedLoss on gfx1250 (MI455X).
// p = sigmoid(pred); all six losses reduce to:
//   * Gram matrix G[c1][c2] = sum_n f_n[c1]*f_n[c2] over N=134400 samples,
//     f = [q_0..q_79, r_0..r_19, 1, pad..] (112 feats), q=p^3, r=(1-p)^3
//   * per-class global max of p for classes 10..79
// Gram via V_WMMA_F32_16X16X4_F32 (28 upper-tri 16x16 tiles, 4/wave, 7 waves).
// Staging via GLOBAL_LOAD_ASYNC_TO_LDS_B128 + s_wait_asynccnt, double-buffered.
// ---------------------------------------------------------------------------

typedef __attribute__((ext_vector_type(2))) float v2f;
typedef __attribute__((ext_vector_type(8))) float v8f;

#define NSAMP   134400            // 16 * 8400
#define CHUNK   56                // samples per LDS pass (56*80*16B = 5 b128/thread)
#define NCHUNK  (NSAMP / CHUNK)   // 2400
#define NCLS    80
#define NFEAT   112               // 7 tiles of 16
#define GRAMSZ  (NFEAT * NFEAT)   // 12544
#define NGRIDWG 240               // 2400 / 240 = 10 chunks per WG
#define NTHREADS 224
#define F4_PER_CHUNK (CHUNK * NCLS / 4)   // 1120 = 5 * 224

// upper-triangular tile-pair list (ti <= tj), 28 pairs, 4 per wave
__constant__ int c_TI[28] = {0,0,0,0, 0,0,0,1, 1,1,1,1, 1,2,2,2,
                             2,2,3,3, 3,3,4,4, 4,5,5,6};
__constant__ int c_TJ[28] = {0,1,2,3, 4,5,6,1, 2,3,4,5, 6,2,3,4,
                             5,6,3,4, 5,6,4,5, 6,5,6,6};

__device__ __forceinline__ float sigm(float x) {
    return 1.0f / (1.0f + __expf(-x));
}

// Issue 5 async b128 copies (global -> LDS) per thread; wave-uniform count.
__device__ __forceinline__ void stage_async(const float4* __restrict__ gsrc,
                                            float4* lds4, int tid) {
#pragma unroll
    for (int k = 0; k < 5; ++k) {
        int i = tid + k * NTHREADS;
        // low 32 bits of the flat shared-aperture address == LDS byte offset
        unsigned ldsOff = (unsigned)(uintptr_t)(lds4 + i);
        unsigned long long ga = (unsigned long long)(uintptr_t)(gsrc + i);
        asm volatile("global_load_async_to_lds_b128 %0, %1, off"
                     :: "v"(ldsOff), "v"(ga)
                     : "memory");
    }
}

__global__ void kb_init(float* ws) {
    int i = blockIdx.x * blockDim.x + threadIdx.x;
    if (i < GRAMSZ + NCLS) ws[i] = 0.0f;   // 0.0f bits == int 0 for max slots
}

__global__ void __launch_bounds__(NTHREADS)
kb_gram(const float* __restrict__ pred, float* __restrict__ gram,
        int* __restrict__ maxBits) {
    __shared__ float raw[2][CHUNK * NCLS];   // 2 x 17.5 KB, double-buffered
    __shared__ float feat[CHUNK * NFEAT];    // 24.5 KB

    const int tid  = threadIdx.x;
    const int lane = tid & 31;
    const int w    = tid >> 5;             // wave 0..6
    const int koff = (lane >> 4) << 1;     // lanes 0-15 -> K{0,1}; 16-31 -> K{2,3}
    const int col  = lane & 15;

    int ti[4], tj[4];
#pragma unroll
    for (int j = 0; j < 4; ++j) { ti[j] = c_TI[4 * w + j]; tj[j] = c_TJ[4 * w + j]; }

    v8f vz = {};
    v8f acc[4];
#pragma unroll
    for (int j = 0; j < 4; ++j) acc[j] = vz;

    const int jb = (tid % 7) * 16;         // class block is stride-invariant (224%7==0)
    float pmax[16];
#pragma unroll
    for (int j = 0; j < 16; ++j) pmax[j] = 0.0f;

    // prologue: kick off first chunk into buffer 0
    const int chunk0 = blockIdx.x;
    stage_async((const float4*)(pred + (size_t)chunk0 * (CHUNK * NCLS)),
                (float4*)raw[0], tid);

    int buf = 0;
    for (int chunk = chunk0; chunk < NCHUNK; chunk += gridDim.x) {
        // ---- issue next chunk's async copy into the other buffer ----
        int nxt = chunk + gridDim.x;
        if (nxt < NCHUNK) {
            stage_async((const float4*)(pred + (size_t)nxt * (CHUNK * NCLS)),
                        (float4*)raw[buf ^ 1], tid);
            // L2 warm-up for the chunk after that
            int nxt2 = nxt + gridDim.x;
            if (nxt2 < NCHUNK)
                __builtin_prefetch(pred + (size_t)nxt2 * (CHUNK * NCLS) + tid * 20, 0, 0);
            // async loads complete in order: <=5 outstanding means chunk c done
            asm volatile("s_wait_asynccnt 0x5" ::: "memory");
        } else {
            asm volatile("s_wait_asynccnt 0x0" ::: "memory");
        }
        __syncthreads();

        // ---- feature pass: q = p^3, r = (1-p)^3, const-1, pad; track max p ----
        for (int p4 = tid; p4 < CHUNK * 7; p4 += NTHREADS) {
            int s = p4 / 7;
            const float* rs = &raw[buf][s * NCLS];
            float* fo = &feat[s * NFEAT + jb];
#pragma unroll
            for (int jj = 0; jj < 16; ++jj) {
                int j = jb + jj;
                float val;
                if (j < 80) {
                    float p = sigm(rs[j]);
                    val = p * p * p;
                    if (j >= 10) pmax[jj] = fmaxf(pmax[jj], p);
                } else if (j < 100) {
                    float p = sigm(rs[j - 80]);
                    float r = 1.0f - p;
                    val = r * r * r;
                } else if (j == 100) {
                    val = 1.0f;
                } else {
                    val = 0.0f;
                }
                fo[jj] = val;
            }
        }
        __syncthreads();

        // ---- Gram accumulation: 14 K-groups of 4 samples, 4 tiles per wave ----
#pragma unroll 1
        for (int g = 0; g < CHUNK / 4; ++g) {
            const float* f0 = &feat[(g * 4 + koff) * NFEAT];
            const float* f1 = f0 + NFEAT;
#pragma unroll
            for (int j = 0; j < 4; ++j) {
                v2f a, b;
                a.x = f0[ti[j] * 16 + col];
                a.y = f1[ti[j] * 16 + col];
                b.x = f0[tj[j] * 16 + col];
                b.y = f1[tj[j] * 16 + col];
                acc[j] = __builtin_amdgcn_wmma_f32_16x16x4_f32(
                    false, a, false, b, (short)0, acc[j], false, false);
            }
        }
        __syncthreads();
        buf ^= 1;
    }

    // ---- flush: one atomic fadd per accumulator element ----
    const int mrow = (lane >> 4) * 8;      // D layout: vgpr v, lanes>=16 -> M=v+8
#pragma unroll
    for (int j = 0; j < 4; ++j) {
#pragma unroll
        for (int v = 0; v < 8; ++v) {
            int m = ti[j] * 16 + v + mrow;
            int n = tj[j] * 16 + col;
            __hip_atomic_fetch_add(&gram[m * NFEAT + n], acc[j][v],
                                   __ATOMIC_RELAXED, __HIP_MEMORY_SCOPE_AGENT);
        }
    }
    // per-class max of p (positive floats: int-bit compare is order-preserving)
#pragma unroll
    for (int jj = 0; jj < 16; ++jj) {
        int j = jb + jj;
        if (j >= 10 && j < 80)
            atomicMax(&maxBits[j], __float_as_int(pmax[jj]));
    }
}

// Gram is stored upper-tile-triangular: swap so tile(c1) <= tile(c2)
__device__ __forceinline__ float gramAt(const float* g, int c1, int c2) {
    if ((c1 >> 4) > (c2 >> 4)) { int t = c1; c1 = c2; c2 = t; }
    return g[c1 * NFEAT + c2];
}

__global__ void kb_final(const float* __restrict__ gram,
                         const int* __restrict__ maxBits,
                         float* __restrict__ out) {
    __shared__ float ssum;
    if (threadIdx.x == 0) ssum = 0.0f;
    __syncthreads();

    const float invN = 1.0f / (float)NSAMP;
    float local = 0.0f;
    // 870 flat terms: [0,420) SE, [420,720) CE, [720,790) G, [790,850) D,
    //                 [850,860) S, [860,870) C
    for (int t = threadIdx.x; t < 870; t += blockDim.x) {
        float term = 0.0f;
        if (t < 420) {                       // SE: 10 rows x 7 x 6 off-diag
            int i = t / 42, rem = t % 42, t1 = rem / 6, k2 = rem % 6;
            int t2 = k2 + (k2 >= t1 ? 1 : 0);
            int c1 = 10 + 7 * i + t1, c2 = 10 + 7 * i + t2;
            term = cbrtf(gramAt(gram, c1, c2) * invN) * (1.0f / (6.0f * 70.0f));
        } else if (t < 720) {                // CE: 10 rows x 6 x 5 off-diag
            int u = t - 420;
            int i = u / 30, rem = u % 30, t1 = rem / 5, k2 = rem % 5;
            int t2 = k2 + (k2 >= t1 ? 1 : 0);
            int c1 = 20 + 6 * i + t1, c2 = 20 + 6 * i + t2;
            term = cbrtf(gramAt(gram, c1, c2) * invN) * (1.0f / (5.0f * 60.0f));
        } else if (t < 790) {                // G: q_{10+k} * r_{k/7}
            int k = t - 720;
            term = cbrtf(gramAt(gram, 10 + k, 80 + k / 7) * invN) * (1.0f / 70.0f);
        } else if (t < 850) {                // D: q_{20+k} * r_{10+k/6}
            int k = t - 790;
            term = cbrtf(gramAt(gram, 20 + k, 90 + k / 6) * invN) * (1.0f / 60.0f);
        } else if (t < 860) {                // S: E[q_i]^(1/3) * (1 - m_i)
            int i = t - 850;
            float m = 0.0f;
#pragma unroll
            for (int tt = 0; tt < 7; ++tt)
                m = fmaxf(m, __int_as_float(maxBits[10 + 7 * i + tt]));
            term = cbrtf(gramAt(gram, i, 100) * invN) * (1.0f - m) * 0.1f;
        } else {                             // C: E[q_{10+i}]^(1/3) * (1 - m'_i)
            int i = t - 860;
            float m = 0.0f;
#pragma unroll
            for (int tt = 0; tt < 6; ++tt)
                m = fmaxf(m, __int_as_float(maxBits[20 + 6 * i + tt]));
            term = cbrtf(gramAt(gram, 10 + i, 100) * invN) * (1.0f - m) * 0.1f;
        }
        local += term;
    }
    atomicAdd(&ssum, local);
    __syncthreads();
    if (threadIdx.x == 0) out[0] = ssum;
}

extern "C" void kernel_launch(void* const* d_in, const int* in_sizes, int n_in,
                              void* d_out, int out_size, void* d_ws, size_t ws_size,
                              hipStream_t stream) {
    (void)in_sizes; (void)n_in; (void)out_size; (void)ws_size;
    const float* pred = (const float*)d_in[0];   // target_scores (d_in[1]) unused
    float* gram = (float*)d_ws;
    int* maxBits = (int*)((float*)d_ws + GRAMSZ);
    float* out = (float*)d_out;

    kb_init<<<(GRAMSZ + NCLS + 255) / 256, 256, 0, stream>>>((float*)d_ws);
    kb_gram<<<NGRIDWG, NTHREADS, 0, stream>>>(pred, gram, maxBits);
    kb_final<<<1, 256, 0, stream>>>(gram, maxBits, out);
}